// DynamicallyComposedMultiHeadAttention_14147622273585
// MI455X (gfx1250) — compile-verified
//
#include <hip/hip_runtime.h>
#include <hip/hip_bf16.h>
#include <math.h>

// ---------------------------------------------------------------------------
// DCMHA for MI455X (gfx1250, wave32). Full-fp32 pipeline: the workload is
// bandwidth-bound (>=205MB mandatory output stores vs ~9 GFLOP of GEMMs), so
// we use V_WMMA_F32_16X16X4_F32 everywhere: tensor-core path, zero precision
// loss.
// ---------------------------------------------------------------------------

typedef float v2f __attribute__((ext_vector_type(2)));
typedef float v8f __attribute__((ext_vector_type(8)));

#define NB 2
#define NH 8
#define TS 1024          // T == S
#define DM 512
#define HD 64
#define SC 128           // s-chunk processed per iteration

static __device__ __forceinline__ v8f wmma4(v2f a, v2f b, v8f c) {
  // D = A(16x4 f32) * B(4x16 f32) + C(16x16 f32)
  return __builtin_amdgcn_wmma_f32_16x16x4_f32(false, a, false, b, (short)0, c,
                                               false, false);
}

// qh[b,h,t,:] / kh[b,h,s,:] are 64 CONTIGUOUS floats of qp/kp (torch .view):
static __device__ __forceinline__ size_t headrow(int b, int h, int t) {
  return ((size_t)b * TS + (size_t)(h * 128 + (t >> 3))) * DM +
         (size_t)(t & 7) * HD;
}

static __device__ __forceinline__ float gelu_exact(float x) {
  return 0.5f * x * (1.0f + erff(x * 0.7071067811865476f));
}

// Online softmax update / merge
static __device__ __forceinline__ void oupd(float x, float& m, float& l) {
  float nm = fmaxf(m, x);
  l = l * __expf(m - nm) + __expf(x - nm);
  m = nm;
}
static __device__ __forceinline__ void omerge(float& m, float& l, float m2,
                                              float l2) {
  float nm = fmaxf(m, m2);
  l = l * __expf(m - nm) + l2 * __expf(m2 - nm);
  m = nm;
}

// compose at one (t,s): o[h] = a[h]*(1+gq[h]+gk[h]) + sum_r dq2[r][h]*<a,dq1[r]>
//                              + sum_r dk2[r][h]*<a,dk1[r]>
// param layout (40 floats): [0..15]=dw1 (r*8+h), [16..31]=dw2, [32..39]=gate
static __device__ __forceinline__ void compose8(const float* a, const float* qd,
                                                const float* kd, float* o) {
  float hq0 = 0.f, hq1 = 0.f, hk0 = 0.f, hk1 = 0.f;
#pragma unroll
  for (int i = 0; i < 8; ++i) {
    float ai = a[i];
    hq0 += ai * qd[i];
    hq1 += ai * qd[8 + i];
    hk0 += ai * kd[i];
    hk1 += ai * kd[8 + i];
  }
#pragma unroll
  for (int h = 0; h < 8; ++h) {
    o[h] = a[h] * (1.f + qd[32 + h] + kd[32 + h]) + qd[16 + h] * hq0 +
           qd[24 + h] * hq1 + kd[16 + h] * hk0 + kd[24 + h] * hk1;
  }
}

// ---------------------------------------------------------------------------
// Kernel 1: Y[2048,512] = X[2048,512] @ W[512,512] + bias   (WMMA fp32)
// grid.x = 128 (16-row tiles), 256 threads = 8 waves, wave w -> cols [w*64..)
// ---------------------------------------------------------------------------
__global__ void k_proj(const float* __restrict__ X, const float* __restrict__ W,
                       const float* __restrict__ bias, float* __restrict__ Y) {
  __shared__ float Xs[16][DM];  // 32 KB
  const int tid = threadIdx.x;
  const int row0 = blockIdx.x * 16;

  const float4* Xv = (const float4*)(X + (size_t)row0 * DM);
  float4* Xsv = (float4*)&Xs[0][0];
  for (int i = tid; i < 16 * (DM / 4); i += 256) Xsv[i] = Xv[i];
  __syncthreads();

  const int wave = tid >> 5, lane = tid & 31;
  const int mn = lane & 15, half = lane >> 4;  // A row / B col ; K sub-pair

  v8f c[4] = {};
  for (int ks = 0; ks < DM / 4; ++ks) {
    const int k0 = ks * 4 + half * 2;
    v2f a;
    a.x = Xs[mn][k0];
    a.y = Xs[mn][k0 + 1];
#pragma unroll
    for (int nt = 0; nt < 4; ++nt) {
      const int n = wave * 64 + nt * 16 + mn;
      v2f bb;
      bb.x = W[(size_t)k0 * DM + n];
      bb.y = W[(size_t)(k0 + 1) * DM + n];
      c[nt] = wmma4(a, bb, c[nt]);
    }
  }
#pragma unroll
  for (int nt = 0; nt < 4; ++nt) {
    const int n = wave * 64 + nt * 16 + mn;
    const float bv = bias[n];
#pragma unroll
    for (int v = 0; v < 8; ++v) {
      const int r = v + half * 8;  // C layout: vgpr v -> row v / v+8
      Y[(size_t)(row0 + r) * DM + n] = c[nt][v] + bv;
    }
  }
}

// ---------------------------------------------------------------------------
// Kernel 2: per-row small params.  dw = gelu(x@P1)@P2 [32], g = tanh(x@Pg) [8]
// for both pre & post sets. wave per row, lane j = output column.
// ---------------------------------------------------------------------------
__global__ void k_small(const float* __restrict__ X,
                        const float* __restrict__ p1a,
                        const float* __restrict__ p2a,
                        const float* __restrict__ pga,
                        const float* __restrict__ p1b,
                        const float* __restrict__ p2b,
                        const float* __restrict__ pgb, float* __restrict__ dwa,
                        float* __restrict__ dwb, float* __restrict__ ga,
                        float* __restrict__ gb) {
  const int wave = threadIdx.x >> 5, lane = threadIdx.x & 31;
  const size_t row = (size_t)blockIdx.x * 8 + wave;
  const float* x = X + row * DM;

  float ta = 0.f, tb = 0.f, gaa = 0.f, gab = 0.f;
  for (int d = 0; d < DM; ++d) {
    const float xv = x[d];
    ta += xv * p1a[d * 32 + lane];
    tb += xv * p1b[d * 32 + lane];
    if (lane < 8) {
      gaa += xv * pga[d * 8 + lane];
      gab += xv * pgb[d * 8 + lane];
    }
  }
  const float gea = gelu_exact(ta), geb = gelu_exact(tb);
  float oa = 0.f, ob = 0.f;
#pragma unroll
  for (int i = 0; i < 32; ++i) {
    const float gi = __shfl(gea, i, 32);
    const float gj = __shfl(geb, i, 32);
    oa += gi * p2a[i * 32 + lane];
    ob += gj * p2b[i * 32 + lane];
  }
  dwa[row * 32 + lane] = oa;
  dwb[row * 32 + lane] = ob;
  if (lane < 8) {
    ga[row * 8 + lane] = tanhf(gaa);
    gb[row * 8 + lane] = tanhf(gab);
  }
}

// ---------------------------------------------------------------------------
// Kernel 3: softmax stats (m,l) for logits rows and pre-composed rows.
// grid.x = 128 : (b, 16-row t-tile). 8 waves = 8 heads compute logits via
// WMMA into the attn tile in LDS, laid out [t][s][h]; 256 compose threads do the 8-vector compose and
// keep online (m,l) per (h,t).
// ---------------------------------------------------------------------------
__global__ void k_stats(const float* __restrict__ qp,
                        const float* __restrict__ kp,
                        const float* __restrict__ dq_pre,
                        const float* __restrict__ gq_pre,
                        const float* __restrict__ dk_pre,
                        const float* __restrict__ gk_pre,
                        float* __restrict__ m0w, float* __restrict__ l0w,
                        float* __restrict__ m1w, float* __restrict__ l1w) {
  extern __shared__ float smem[];
  float* attn = smem;                      // 16*SC*8
  float* kpar = attn + 16 * SC * 8;        // SC*40
  float* qpar = kpar + SC * 40;            // 16*40
  float* red = qpar + 16 * 40;             // 8*16*16*4

  const int tid = threadIdx.x;
  const int b = blockIdx.x >> 6;
  const int t0 = (blockIdx.x & 63) * 16;
  const int wave = tid >> 5, lane = tid & 31;
  const int mn = lane & 15, half = lane >> 4;
  const int ct = tid >> 4, cs = tid & 15;  // compose thread: t, s-group

  // stage per-t q params
  for (int i = tid; i < 16 * 40; i += 256) {
    const int t = i / 40, j = i % 40;
    const size_t r = (size_t)b * TS + t0 + t;
    qpar[i] = (j < 32) ? dq_pre[r * 32 + j] : gq_pre[r * 8 + (j - 32)];
  }
  // A tile (qh) for this wave's head: 16 rows x 64, fixed for whole block
  const float* qr = qp + headrow(b, wave, t0 + mn);
  v2f areg[16];
#pragma unroll
  for (int ks = 0; ks < 16; ++ks) {
    areg[ks].x = qr[4 * ks + 2 * half];
    areg[ks].y = qr[4 * ks + 2 * half + 1];
  }

  float pm0[8], pl0[8], pm1[8], pl1[8];
#pragma unroll
  for (int h = 0; h < 8; ++h) {
    pm0[h] = -3.0e38f; pl0[h] = 0.f; pm1[h] = -3.0e38f; pl1[h] = 0.f;
  }

  for (int sc = 0; sc < TS / SC; ++sc) {
    const int s0c = sc * SC;
    __syncthreads();
    // stage per-s k params for this chunk
    for (int i = tid; i < SC * 40; i += 256) {
      const int s = i / 40, j = i % 40;
      const size_t r = (size_t)b * TS + s0c + s;
      kpar[i] = (j < 32) ? dk_pre[r * 32 + j] : gk_pre[r * 8 + (j - 32)];
    }
    // logits: wave = head, 8 s-tiles of 16
    for (int st = 0; st < SC / 16; ++st) {
      const int s = s0c + st * 16 + mn;
      const float* kr = kp + headrow(b, wave, s);
      v8f c = {};
#pragma unroll
      for (int ks = 0; ks < 16; ++ks) {
        v2f bb;
        bb.x = kr[4 * ks + 2 * half];
        bb.y = kr[4 * ks + 2 * half + 1];
        c = wmma4(areg[ks], bb, c);
      }
#pragma unroll
      for (int v = 0; v < 8; ++v) {
        const int mr = v + half * 8;
        attn[((size_t)mr * SC + st * 16 + mn) * 8 + wave] = c[v] * 0.125f;
      }
    }
    __syncthreads();
    // compose(pre) + online stats
    for (int j = 0; j < SC / 16; ++j) {
      const int sl = cs + 16 * j;
      const float* av = &attn[((size_t)ct * SC + sl) * 8];
      float a[8], pre[8];
#pragma unroll
      for (int h = 0; h < 8; ++h) a[h] = av[h];
      compose8(a, &qpar[ct * 40], &kpar[sl * 40], pre);
#pragma unroll
      for (int h = 0; h < 8; ++h) {
        oupd(a[h], pm0[h], pl0[h]);
        oupd(pre[h], pm1[h], pl1[h]);
      }
    }
  }
  // reduce 16 s-group partials per (h,t)
#pragma unroll
  for (int h = 0; h < 8; ++h) {
    const int base = ((h * 16 + ct) * 16 + cs) * 4;
    red[base + 0] = pm0[h]; red[base + 1] = pl0[h];
    red[base + 2] = pm1[h]; red[base + 3] = pl1[h];
  }
  __syncthreads();
  if (tid < 128) {
    const int h = tid >> 4, t = tid & 15;
    float m0 = -3.0e38f, l0 = 0.f, m1 = -3.0e38f, l1 = 0.f;
    for (int g = 0; g < 16; ++g) {
      const int base = ((h * 16 + t) * 16 + g) * 4;
      omerge(m0, l0, red[base + 0], red[base + 1]);
      omerge(m1, l1, red[base + 2], red[base + 3]);
    }
    const size_t idx = ((size_t)b * NH + h) * TS + t0 + t;
    m0w[idx] = m0; l0w[idx] = l0; m1w[idx] = m1; l1w[idx] = l1;
  }
}

// ---------------------------------------------------------------------------
// Kernel 4: final pass. Recompute logits (cheap), emit uncomposed / probs /
// post, and accumulate out = post @ vh via WMMA from an LDS post tile.
// ---------------------------------------------------------------------------
__global__ void k_final(
    const float* __restrict__ qp, const float* __restrict__ kp,
    const float* __restrict__ vp, const float* __restrict__ dq_pre,
    const float* __restrict__ gq_pre, const float* __restrict__ dk_pre,
    const float* __restrict__ gk_pre, const float* __restrict__ dq_post,
    const float* __restrict__ gq_post, const float* __restrict__ dk_post,
    const float* __restrict__ gk_post, const float* __restrict__ m0w,
    const float* __restrict__ l0w, const float* __restrict__ m1w,
    const float* __restrict__ l1w, float* __restrict__ outp,
    float* __restrict__ unc, float* __restrict__ probs,
    float* __restrict__ postT) {
  extern __shared__ float smem[];
  float* attn = smem;                       // 16*SC*8
  float* postb = attn + 16 * SC * 8;        // 8*16*SC  ([h][t][s])
  float* kpar = postb + 8 * 16 * SC;        // SC*80 (pre 0..39, post 40..79)
  float* qpar = kpar + SC * 80;             // 16*80
  float* stat = qpar + 16 * 80;             // 8*16*4 : m0, 1/l0, m1, 1/l1

  const int tid = threadIdx.x;
  const int b = blockIdx.x >> 6;
  const int t0 = (blockIdx.x & 63) * 16;
  const int wave = tid >> 5, lane = tid & 31;
  const int mn = lane & 15, half = lane >> 4;
  const int ct = tid >> 4, cs = tid & 15;

  for (int i = tid; i < 16 * 80; i += 256) {
    const int t = i / 80, j = i % 80;
    const size_t r = (size_t)b * TS + t0 + t;
    float v;
    if (j < 32)      v = dq_pre[r * 32 + j];
    else if (j < 40) v = gq_pre[r * 8 + (j - 32)];
    else if (j < 72) v = dq_post[r * 32 + (j - 40)];
    else             v = gq_post[r * 8 + (j - 72)];
    qpar[i] = v;
  }
  if (tid < 128) {
    const int h = tid >> 4, t = tid & 15;
    const size_t idx = ((size_t)b * NH + h) * TS + t0 + t;
    stat[tid * 4 + 0] = m0w[idx];
    stat[tid * 4 + 1] = 1.f / l0w[idx];
    stat[tid * 4 + 2] = m1w[idx];
    stat[tid * 4 + 3] = 1.f / l1w[idx];
  }

  const float* qr = qp + headrow(b, wave, t0 + mn);
  v2f areg[16];
#pragma unroll
  for (int ks = 0; ks < 16; ++ks) {
    areg[ks].x = qr[4 * ks + 2 * half];
    areg[ks].y = qr[4 * ks + 2 * half + 1];
  }

  v8f acc[4] = {};  // out tile: 16 t x 64 d (head = wave)

  for (int sc = 0; sc < TS / SC; ++sc) {
    const int s0c = sc * SC;
    __syncthreads();
    for (int i = tid; i < SC * 80; i += 256) {
      const int s = i / 80, j = i % 80;
      const size_t r = (size_t)b * TS + s0c + s;
      float v;
      if (j < 32)      v = dk_pre[r * 32 + j];
      else if (j < 40) v = gk_pre[r * 8 + (j - 32)];
      else if (j < 72) v = dk_post[r * 32 + (j - 40)];
      else             v = gk_post[r * 8 + (j - 72)];
      kpar[i] = v;
    }
    if (sc + 1 < TS / SC)  // pull next chunk's K rows toward the caches
      __builtin_prefetch(kp + headrow(b, wave, s0c + SC + mn), 0, 0);
    // logits via WMMA -> attn[t][s][h]
    for (int st = 0; st < SC / 16; ++st) {
      const int s = s0c + st * 16 + mn;
      const float* kr = kp + headrow(b, wave, s);
      v8f c = {};
#pragma unroll
      for (int ks = 0; ks < 16; ++ks) {
        v2f bb;
        bb.x = kr[4 * ks + 2 * half];
        bb.y = kr[4 * ks + 2 * half + 1];
        c = wmma4(areg[ks], bb, c);
      }
#pragma unroll
      for (int v = 0; v < 8; ++v) {
        const int mr = v + half * 8;
        attn[((size_t)mr * SC + st * 16 + mn) * 8 + wave] = c[v] * 0.125f;
      }
    }
    __syncthreads();
    // elementwise: uncomposed / pre-compose / probs / post-compose
    for (int j = 0; j < SC / 16; ++j) {
      const int sl = cs + 16 * j;
      const int sg = s0c + sl;
      const float* av = &attn[((size_t)ct * SC + sl) * 8];
      float a[8], pre[8], p[8], o[8];
#pragma unroll
      for (int h = 0; h < 8; ++h) a[h] = av[h];
      compose8(a, &qpar[ct * 80], &kpar[sl * 80], pre);
#pragma unroll
      for (int h = 0; h < 8; ++h) {
        const float* s4 = &stat[(h * 16 + ct) * 4];
        const size_t gidx =
            ((size_t)b * NH + h) * ((size_t)TS * TS) + (size_t)(t0 + ct) * TS + sg;
        unc[gidx] = __expf(a[h] - s4[0]) * s4[1];
        p[h] = __expf(pre[h] - s4[2]) * s4[3];
        probs[gidx] = p[h];
      }
      compose8(p, &qpar[ct * 80 + 40], &kpar[sl * 80 + 40], o);
#pragma unroll
      for (int h = 0; h < 8; ++h) {
        const size_t gidx =
            ((size_t)b * NH + h) * ((size_t)TS * TS) + (size_t)(t0 + ct) * TS + sg;
        postT[gidx] = o[h];
        postb[((size_t)h * 16 + ct) * SC + sl] = o[h];
      }
    }
    __syncthreads();
    // out += post[16 x SC] @ vh[SC x 64]   (A from LDS, B from vp)
    for (int ks = 0; ks < SC / 4; ++ks) {
      const int k0 = ks * 4 + 2 * half;
      v2f a2;
      a2.x = postb[((size_t)wave * 16 + mn) * SC + k0];
      a2.y = postb[((size_t)wave * 16 + mn) * SC + k0 + 1];
      const int sg = s0c + k0;
      const float* v0 = vp + headrow(b, wave, sg);
      const float* v1 = vp + headrow(b, wave, sg + 1);
#pragma unroll
      for (int dt = 0; dt < 4; ++dt) {
        v2f b2;
        b2.x = v0[dt * 16 + mn];
        b2.y = v1[dt * 16 + mn];
        acc[dt] = wmma4(a2, b2, acc[dt]);
      }
    }
  }
  // store out[b, t, h*64 + d]
#pragma unroll
  for (int dt = 0; dt < 4; ++dt) {
#pragma unroll
    for (int v = 0; v < 8; ++v) {
      const int r = v + half * 8;
      outp[((size_t)b * TS + t0 + r) * DM + wave * 64 + dt * 16 + mn] =
          acc[dt][v];
    }
  }
}

// ---------------------------------------------------------------------------
extern "C" void kernel_launch(void* const* d_in, const int* in_sizes, int n_in,
                              void* d_out, int out_size, void* d_ws,
                              size_t ws_size, hipStream_t stream) {
  const float* query  = (const float*)d_in[0];
  const float* key    = (const float*)d_in[1];
  const float* value  = (const float*)d_in[2];
  const float* wq     = (const float*)d_in[3];
  const float* bq     = (const float*)d_in[4];
  const float* wk     = (const float*)d_in[5];
  const float* bk     = (const float*)d_in[6];
  const float* wv     = (const float*)d_in[7];
  const float* bv     = (const float*)d_in[8];
  const float* pre_q1 = (const float*)d_in[9];
  const float* pre_q2 = (const float*)d_in[10];
  const float* pre_k1 = (const float*)d_in[11];
  const float* pre_k2 = (const float*)d_in[12];
  const float* pre_qg = (const float*)d_in[13];
  const float* pre_kg = (const float*)d_in[14];
  const float* post_q1 = (const float*)d_in[15];
  const float* post_q2 = (const float*)d_in[16];
  const float* post_k1 = (const float*)d_in[17];
  const float* post_k2 = (const float*)d_in[18];
  const float* post_qg = (const float*)d_in[19];
  const float* post_kg = (const float*)d_in[20];

  // workspace carve (floats): ~14.2 MB total
  float* ws = (float*)d_ws;
  float* qp = ws;                       // 2*1024*512
  float* kp = qp + 1048576;
  float* vp = kp + 1048576;
  float* dq_pre  = vp + 1048576;        // 2048*32 each
  float* dq_post = dq_pre + 65536;
  float* dk_pre  = dq_post + 65536;
  float* dk_post = dk_pre + 65536;
  float* gq_pre  = dk_post + 65536;     // 2048*8 each
  float* gq_post = gq_pre + 16384;
  float* gk_pre  = gq_post + 16384;
  float* gk_post = gk_pre + 16384;
  float* m0 = gk_post + 16384;          // 2*8*1024 each
  float* l0 = m0 + 16384;
  float* m1 = l0 + 16384;
  float* l1 = m1 + 16384;

  // output carve: (out, uncomposed, probs, post)
  float* outp  = (float*)d_out;
  float* unc   = outp + 1048576;
  float* probs = unc + 16777216;
  float* postT = probs + 16777216;

  k_proj<<<128, 256, 0, stream>>>(query, wq, bq, qp);
  k_proj<<<128, 256, 0, stream>>>(key, wk, bk, kp);
  k_proj<<<128, 256, 0, stream>>>(value, wv, bv, vp);

  k_small<<<256, 256, 0, stream>>>(query, pre_q1, pre_q2, pre_qg, post_q1,
                                   post_q2, post_qg, dq_pre, dq_post, gq_pre,
                                   gq_post);
  k_small<<<256, 256, 0, stream>>>(key, pre_k1, pre_k2, pre_kg, post_k1,
                                   post_k2, post_kg, dk_pre, dk_post, gk_pre,
                                   gk_post);

  const size_t smem3 =
      (size_t)(16 * SC * 8 + SC * 40 + 16 * 40 + 8 * 16 * 16 * 4) * 4;
  k_stats<<<128, 256, smem3, stream>>>(qp, kp, dq_pre, gq_pre, dk_pre, gk_pre,
                                       m0, l0, m1, l1);

  const size_t smem4 =
      (size_t)(16 * SC * 8 + 8 * 16 * SC + SC * 80 + 16 * 80 + 8 * 16 * 4) * 4;
  k_final<<<128, 256, smem4, stream>>>(qp, kp, vp, dq_pre, gq_pre, dk_pre,
                                       gk_pre, dq_post, gq_post, dk_post,
                                       gk_post, m0, l0, m1, l1, outp, unc,
                                       probs, postT);
}